// GATLayer_14972255994477
// MI455X (gfx1250) — compile-verified
//
#include <hip/hip_runtime.h>

typedef __attribute__((ext_vector_type(2))) float v2f;
typedef __attribute__((ext_vector_type(8))) float v8f;

#define IN_DIM  128
#define OUT_DIM 64

// ---------------------------------------------------------------- zero init
__global__ __launch_bounds__(256) void zero_kernel(float* __restrict__ out,
                                                   int* __restrict__ deg,
                                                   int nOut, int nDeg) {
  int i = blockIdx.x * blockDim.x + threadIdx.x;
  int stride = gridDim.x * blockDim.x;
  for (int j = i; j < nOut; j += stride) out[j] = 0.0f;
  for (int j = i; j < nDeg; j += stride) deg[j] = 0;
}

// ------------------------------------------------------------ in-degree hist
__global__ __launch_bounds__(256) void degree_kernel(const int* __restrict__ dst,
                                                     int* __restrict__ deg, int E) {
  int e = blockIdx.x * blockDim.x + threadIdx.x;
  if (e < E) atomicAdd(&deg[dst[e]], 1);
}

// ------------------------------------------------- z = h @ Wfc^T  (FP32 WMMA)
// One wave computes a 16-row stripe of z across all 64 output columns using
// V_WMMA_F32_16X16X4_F32:  D(16x16) = A(16x4) * B(4x16) + C.
// A lane layout (ISA 7.12.2, 32-bit A 16x4): lane l<16 -> M=l, VGPR0=K0,VGPR1=K1;
//                                            lane l>=16 -> M=l-16, VGPR0=K2,VGPR1=K3.
// B lane layout (row striped across lanes): VGPR0 = rows K0(lanes0-15)/K2(16-31),
//                                           VGPR1 = rows K1/K3, N = lane&15.
__global__ __launch_bounds__(256) void gemm_wmma_kernel(const float* __restrict__ h,
                                                        const float* __restrict__ W,
                                                        float* __restrict__ z,
                                                        int nTilesM, int N) {
  int wave = (int)((blockIdx.x * blockDim.x + threadIdx.x) >> 5);
  if (wave >= nTilesM) return;                 // uniform per wave: EXEC stays all-1s
  int lane = threadIdx.x & 31;
  int half = lane >> 4;                        // 0 or 1
  int lr   = lane & 15;                        // 0..15

  int m0 = wave * 16;
  int rowA = m0 + lr;
  if (rowA > N - 1) rowA = N - 1;              // clamp (keeps all lanes active)
  const float* aPtr  = h + (size_t)rowA * IN_DIM + 2 * half;
  const float* bPtr0 = W + (size_t)(0 * 16 + lr) * IN_DIM + 2 * half;
  const float* bPtr1 = W + (size_t)(1 * 16 + lr) * IN_DIM + 2 * half;
  const float* bPtr2 = W + (size_t)(2 * 16 + lr) * IN_DIM + 2 * half;
  const float* bPtr3 = W + (size_t)(3 * 16 + lr) * IN_DIM + 2 * half;

  v8f acc0 = {}, acc1 = {}, acc2 = {}, acc3 = {};

#pragma unroll 4
  for (int k = 0; k < IN_DIM; k += 4) {
    v2f a  = *(const v2f*)(aPtr + k);
    v2f b0 = *(const v2f*)(bPtr0 + k);
    v2f b1 = *(const v2f*)(bPtr1 + k);
    v2f b2 = *(const v2f*)(bPtr2 + k);
    v2f b3 = *(const v2f*)(bPtr3 + k);
    acc0 = __builtin_amdgcn_wmma_f32_16x16x4_f32(false, a, false, b0, (short)0, acc0, false, false);
    acc1 = __builtin_amdgcn_wmma_f32_16x16x4_f32(false, a, false, b1, (short)0, acc1, false, false);
    acc2 = __builtin_amdgcn_wmma_f32_16x16x4_f32(false, a, false, b2, (short)0, acc2, false, false);
    acc3 = __builtin_amdgcn_wmma_f32_16x16x4_f32(false, a, false, b3, (short)0, acc3, false, false);
  }

  // D layout: lane l, vgpr i -> row m0 + i + 8*half, col 16*t + (l&15)
#pragma unroll
  for (int i = 0; i < 8; ++i) {
    int m = m0 + i + 8 * half;
    if (m < N) {
      float* zr = z + (size_t)m * OUT_DIM;
      zr[0 * 16 + lr] = acc0[i];
      zr[1 * 16 + lr] = acc1[i];
      zr[2 * 16 + lr] = acc2[i];
      zr[3 * 16 + lr] = acc3[i];
    }
  }
}

// --------------------------------------------- edge scatter: out[dst] += z[src]
// One wave per edge; each lane handles 2 consecutive columns (float2 gather,
// two global_atomic_add_f32). z and out both fit in the 192MB L2.
__global__ __launch_bounds__(256) void edge_kernel(const int* __restrict__ src,
                                                   const int* __restrict__ dst,
                                                   const float* __restrict__ z,
                                                   float* __restrict__ out, int E) {
  int e = blockIdx.x * 8 + (int)(threadIdx.x >> 5);
  if (e >= E) return;
  int lane = threadIdx.x & 31;
  int s = src[e];
  int d = dst[e];
  v2f v = *(const v2f*)(z + (size_t)s * OUT_DIM + 2 * lane);
  atomicAdd(out + (size_t)d * OUT_DIM + 2 * lane + 0, v.x);
  atomicAdd(out + (size_t)d * OUT_DIM + 2 * lane + 1, v.y);
}

// ------------------------------------------------------- row scale by 1/deg
__global__ __launch_bounds__(256) void scale_kernel(float* __restrict__ out,
                                                    const int* __restrict__ deg, int N) {
  int idx = blockIdx.x * blockDim.x + threadIdx.x;
  if (idx >= N * OUT_DIM) return;
  int v = idx / OUT_DIM;
  int d = deg[v];
  float s = (d > 0) ? (1.0f / (float)d) : 0.0f;
  out[idx] *= s;
}

extern "C" void kernel_launch(void* const* d_in, const int* in_sizes, int n_in,
                              void* d_out, int out_size, void* d_ws, size_t ws_size,
                              hipStream_t stream) {
  const float* h   = (const float*)d_in[0];
  // d_in[1] = u   : unused (attention output replaced by ones in reference)
  const int*   src = (const int*)d_in[2];
  const int*   dst = (const int*)d_in[3];
  const float* Wfc = (const float*)d_in[4];
  // d_in[5] = W2  : unused

  int N = in_sizes[0] / IN_DIM;
  int E = in_sizes[2];

  float* out = (float*)d_out;
  float* z   = (float*)d_ws;                                      // N*64 floats
  int*   deg = (int*)((char*)d_ws + (size_t)N * OUT_DIM * sizeof(float));

  zero_kernel<<<1024, 256, 0, stream>>>(out, deg, N * OUT_DIM, N);
  degree_kernel<<<(E + 255) / 256, 256, 0, stream>>>(dst, deg, E);

  int nTilesM = (N + 15) / 16;
  int gemmBlocks = (nTilesM * 32 + 255) / 256;
  gemm_wmma_kernel<<<gemmBlocks, 256, 0, stream>>>(h, Wfc, z, nTilesM, N);

  edge_kernel<<<(E + 7) / 8, 256, 0, stream>>>(src, dst, z, out, E);
  scale_kernel<<<(N * OUT_DIM + 255) / 256, 256, 0, stream>>>(out, deg, N);
}